// TimeConv_17291538333993
// MI455X (gfx1250) — compile-verified
//
#include <hip/hip_runtime.h>
#include <math.h>

typedef __attribute__((ext_vector_type(2))) float v2f;
typedef __attribute__((ext_vector_type(8))) float v8f;
typedef __attribute__((ext_vector_type(4))) unsigned int v4u;
typedef __attribute__((ext_vector_type(4))) int v4i;
typedef __attribute__((ext_vector_type(8))) int v8i;

#define TT 32
#define BNN 16384
#define CH 128
#define PI_F 3.14159265358979323846f

// LDS layout (floats): XT[16][5][128] | Wbuf0[32768] | Wbuf1[32768]
#define XT_FLOATS   10240
#define WB_FLOATS   32768
#define W0_BYTEOFF  (XT_FLOATS * 4)                 // 40960
#define WB_BYTES    (WB_FLOATS * 4)                 // 131072
#define LDS_BYTES   ((XT_FLOATS + 2 * WB_FLOATS) * 4)  // 303104 <= 320KB

#if defined(__has_builtin)
#if __has_builtin(__builtin_amdgcn_tensor_load_to_lds)
#define HAVE_TDM 1
#endif
#endif

// ---------------------------------------------------------------------------
// Kernel 0: repack weights (C,C,3) -> wT[m][reim][k/2][o][p], p = k&1.
// This bakes the WMMA B-fragment swizzle in: fragment {W[k][o],W[k+1][o]}
// becomes one aligned 8-byte LDS load. Writes are fully coalesced (dst=tid).
// ---------------------------------------------------------------------------
__global__ __launch_bounds__(256) void fno_wtrans(const float* __restrict__ wr,
                                                  const float* __restrict__ wi,
                                                  float* __restrict__ wT) {
    int tid = blockIdx.x * blockDim.x + threadIdx.x;
    if (tid >= 2 * 3 * CH * CH) return;
    int p    = tid & 1;
    int o    = (tid >> 1) & 127;
    int k2   = (tid >> 8) & 63;
    int reim = (tid >> 14) & 1;
    int m    = tid >> 15;
    int i    = 2 * k2 + p;         // input channel == K index
    const float* src = reim ? wi : wr;
    wT[tid] = src[(i * CH + o) * 3 + m];
}

#ifdef HAVE_TDM
// Issue a TDM DMA of one mode's 32768-float weight block (global -> LDS).
// 1-D tensor: data_size=4B, tensor_dim0 = tile_dim0 = 32768, type=2 ("image").
__device__ __forceinline__ void tdm_load_block(const float* gsrc, unsigned lds_byte_off) {
    unsigned long long ga = (unsigned long long)(const void*)gsrc;
    v4u g0;
    g0[0] = 1u;                                            // count=1, user mode
    g0[1] = lds_byte_off;                                  // lds_addr [63:32]
    g0[2] = (unsigned)(ga & 0xffffffffull);                // global_addr [95:64]
    g0[3] = (unsigned)((ga >> 32) & 0x01ffffffull)         // global_addr [120:96]
            | 0x80000000u;                                 // type=2 in [127:126]
    v8i g1;
    g1[0] = 0x00020000;        // workgroup_mask=0, data_size=2 (4 bytes)
    g1[1] = (int)0x80000000;   // tensor_dim0 = 32768 (low16 @ bits 63:48)
    g1[2] = 0x00010000;        // tensor_dim0 hi=0; tensor_dim1 = 1
    g1[3] = (int)0x80000000;   // tensor_dim1 hi=0; tile_dim0 = 32768
    g1[4] = 0;                 // tile_dim1 = tile_dim2 = 0 (unused)
    g1[5] = 32768;             // tensor_dim0_stride
    g1[6] = 0;
    g1[7] = 0;
    v4i z4 = {0, 0, 0, 0};
#if __clang_major__ >= 23
    v8i z8 = {0, 0, 0, 0, 0, 0, 0, 0};
    __builtin_amdgcn_tensor_load_to_lds(g0, g1, z4, z4, z8, 0);
#else
    __builtin_amdgcn_tensor_load_to_lds(g0, g1, z4, z4, 0);
#endif
}
#endif

// ---------------------------------------------------------------------------
// Main fused kernel: 3-mode DFT -> complex channel mix (f32 WMMA) -> iDFT
// -> residual leaky-relu.  One workgroup (8 waves) per 16 batch rows.
// Weights double-buffered in LDS, streamed in by the Tensor Data Mover.
// ---------------------------------------------------------------------------
__global__ __launch_bounds__(256) void fno_main(const float* __restrict__ h,
                                                const float* __restrict__ wT,
                                                float* __restrict__ out) {
    extern __shared__ float smem[];
    float* XT = smem;                    // 16*5*128
    float* WBUF0 = smem + XT_FLOATS;     // mode weights, buffer 0
    float* WBUF1 = WBUF0 + WB_FLOATS;    // mode weights, buffer 1

    const int tid   = threadIdx.x;
    const int lane  = tid & 31;
    const int wav   = tid >> 5;
    const int laneM = lane & 15;   // row (A/M) or col (B/N) within 16
    const int kh    = lane >> 4;   // lane-half: selects K pair / row+8
    const int o0    = wav * 16;    // this wave's 16 output channels
    const int b0    = blockIdx.x * 16;

    const float w1 = 2.0f * PI_F / 32.0f;

#ifdef HAVE_TDM
    // Kick off DMA of mode-0 and mode-1 weights; overlaps the whole DFT stage.
    if (wav == 0) {
        tdm_load_block(wT + 0 * WB_FLOATS, W0_BYTEOFF + 0 * WB_BYTES);
        tdm_load_block(wT + 1 * WB_FLOATS, W0_BYTEOFF + 1 * WB_BYTES);
    }
#endif

    // ---------------- Stage 1: 3-mode DFT over t ----------------
    // thread owns row b_local = tid/16, channels i0..i0+7 (i0 = (tid%16)*8)
    float x0[8], x1r[8], x1i[8], x2r[8], x2i[8];
#pragma unroll
    for (int j = 0; j < 8; ++j) { x0[j] = 0.f; x1r[j] = 0.f; x1i[j] = 0.f; x2r[j] = 0.f; x2i[j] = 0.f; }

    for (int t = 0; t < TT; ++t) {
        const float4* hp = (const float4*)(h + (size_t)t * BNN * CH + (size_t)b0 * CH);
        if (t + 1 < TT) {   // hint next t-plane of this tile (global_prefetch_b8)
            __builtin_prefetch(h + (size_t)(t + 1) * BNN * CH + (size_t)b0 * CH + tid * 8, 0, 0);
        }
        float4 p0 = hp[tid * 2 + 0];
        float4 p1 = hp[tid * 2 + 1];
        float s1, c1, s2, c2;
        __sincosf(w1 * (float)t, &s1, &c1);
        __sincosf(2.0f * w1 * (float)t, &s2, &c2);
        float v[8] = {p0.x, p0.y, p0.z, p0.w, p1.x, p1.y, p1.z, p1.w};
#pragma unroll
        for (int j = 0; j < 8; ++j) {
            x0[j]  += v[j];
            x1r[j] += v[j] * c1;  x1i[j] -= v[j] * s1;
            x2r[j] += v[j] * c2;  x2i[j] -= v[j] * s2;
        }
    }
    {
        const int bl = tid >> 4;
        const int i0 = (tid & 15) * 8;
#define STORE_COMP(c, a)                                                     \
        do {                                                                 \
            float* p = &XT[((bl) * 5 + (c)) * 128 + i0];                     \
            ((float4*)p)[0] = make_float4(a[0], a[1], a[2], a[3]);           \
            ((float4*)p)[1] = make_float4(a[4], a[5], a[6], a[7]);           \
        } while (0)
        STORE_COMP(0, x0);
        STORE_COMP(1, x1r);
        STORE_COMP(2, x1i);
        STORE_COMP(3, x2r);
        STORE_COMP(4, x2i);
#undef STORE_COMP
    }

    // ---------------- Stage 2: per-mode complex matmul via f32 WMMA --------
    v8f acc0  = {};                 // Y0r  (Y0i never needed: irfft drops DC imag)
    v8f accR1 = {}, accI1 = {};     // Y1r, Y1i
    v8f accR2 = {}, accI2 = {};     // Y2r, Y2i

    for (int m = 0; m < 3; ++m) {
#ifdef HAVE_TDM
        if (wav == 0) {
            if (m == 1) tdm_load_block(wT + 2 * WB_FLOATS, W0_BYTEOFF);  // prefetch mode 2 -> buf0
            if (m < 2) __builtin_amdgcn_s_wait_tensorcnt(1);  // mode m landed, prefetch in flight
            else       __builtin_amdgcn_s_wait_tensorcnt(0);
        }
        __syncthreads();
#else
        __syncthreads();   // previous mode's LDS reads done (also covers XT stores)
        {   // cooperative stage of this mode's Wr|Wi (32768 floats, coalesced)
            const float4* srcw = (const float4*)(wT + m * WB_FLOATS);
            float4* dstw = (float4*)((m & 1) ? WBUF1 : WBUF0);
#pragma unroll
            for (int it = 0; it < 32; ++it) dstw[it * 256 + tid] = srcw[it * 256 + tid];
        }
        __syncthreads();
#endif
        // Pair-swizzled layout: W?[ (k>>1)*256 + o*2 + (k&1) ]
        const float* WR = (m & 1) ? WBUF1 : WBUF0;
        const float* WI = WR + 16384;

        if (m == 0) {
#pragma unroll
            for (int k = 0; k < 128; k += 4) {
                const int krow = k + 2 * kh;   // even
                v2f a = *(const v2f*)&XT[(laneM * 5 + 0) * 128 + krow];
                v2f b = *(const v2f*)&WR[(krow >> 1) * 256 + (o0 + laneM) * 2];
                acc0 = __builtin_amdgcn_wmma_f32_16x16x4_f32(false, a, false, b,
                                                             (short)0, acc0, false, false);
            }
        } else {
            v8f aR = (m == 1) ? accR1 : accR2;
            v8f aI = (m == 1) ? accI1 : accI2;
            const int cr = 2 * m - 1, ci = 2 * m;
#pragma unroll
            for (int k = 0; k < 128; k += 4) {
                const int krow = k + 2 * kh;   // even
                v2f ar = *(const v2f*)&XT[(laneM * 5 + cr) * 128 + krow];
                v2f ai = *(const v2f*)&XT[(laneM * 5 + ci) * 128 + krow];
                v2f br = *(const v2f*)&WR[(krow >> 1) * 256 + (o0 + laneM) * 2];
                v2f bi = *(const v2f*)&WI[(krow >> 1) * 256 + (o0 + laneM) * 2];
                aR = __builtin_amdgcn_wmma_f32_16x16x4_f32(false, ar, false, br,
                                                           (short)0, aR, false, false);
                aI = __builtin_amdgcn_wmma_f32_16x16x4_f32(false, ar, false, bi,
                                                           (short)0, aI, false, false);
                v2f nbi = -bi;   // f32 WMMA has no A/B neg modifier: negate B frag in VALU
                aR = __builtin_amdgcn_wmma_f32_16x16x4_f32(false, ai, false, nbi,
                                                           (short)0, aR, false, false);
                aI = __builtin_amdgcn_wmma_f32_16x16x4_f32(false, ai, false, br,
                                                           (short)0, aI, false, false);
            }
            if (m == 1) { accR1 = aR; accI1 = aI; } else { accR2 = aR; accI2 = aI; }
        }
#ifdef HAVE_TDM
        __syncthreads();   // all reads of this buffer done before TDM may overwrite it
#endif
    }

    // ---------------- Stage 3: iDFT (3 modes) + residual leaky-relu --------
    // D layout: VGPR v, lanes<16 -> row v, lanes>=16 -> row v+8; col = lane%16.
    const int col      = o0 + laneM;
    const int rowBase  = b0 + 8 * kh;
    const size_t plane = (size_t)BNN * CH;
    const float* hbase = h   + (size_t)rowBase * CH + col;
    float*       obase = out + (size_t)rowBase * CH + col;

    for (int t = 0; t < TT; ++t) {
        float s1, c1, s2, c2;
        __sincosf(w1 * (float)t, &s1, &c1);
        __sincosf(2.0f * w1 * (float)t, &s2, &c2);
        const size_t toff = (size_t)t * plane;
#pragma unroll
        for (int v = 0; v < 8; ++v) {
            float y = (acc0[v]
                       + 2.0f * (accR1[v] * c1 - accI1[v] * s1)
                       + 2.0f * (accR2[v] * c2 - accI2[v] * s2)) * (1.0f / 32.0f);
            float hv = hbase[toff + (size_t)v * CH];
            float yl = (y >= 0.0f) ? y : 0.2f * y;
            obase[toff + (size_t)v * CH] = hv + yl;
        }
    }
}

// ---------------------------------------------------------------------------
extern "C" void kernel_launch(void* const* d_in, const int* in_sizes, int n_in,
                              void* d_out, int out_size, void* d_ws, size_t ws_size,
                              hipStream_t stream) {
    const float* h  = (const float*)d_in[0];
    const float* wr = (const float*)d_in[1];
    const float* wi = (const float*)d_in[2];
    float* out = (float*)d_out;
    float* wT  = (float*)d_ws;   // 98304 floats = 384 KB scratch

    fno_wtrans<<<(2 * 3 * CH * CH + 255) / 256, 256, 0, stream>>>(wr, wi, wT);

    fno_main<<<BNN / 16, 256, LDS_BYTES, stream>>>(h, wT, out);
}